// BERT_CRF_39178691674351
// MI455X (gfx1250) — compile-verified
//
#include <hip/hip_runtime.h>
#include <math.h>

typedef __attribute__((ext_vector_type(2))) float v2f;
typedef __attribute__((ext_vector_type(8))) float v8f;

#define Bq 64
#define Sq 512
#define Hq 768
#define Lq 9
#define LP 16              // L padded to one WMMA tile width
#define Mq (Bq * Sq)       // 32768 rows

// ---------------------------------------------------------------------------
// Kernel 1: emissions[m][n] = sum_k hs[m][k] * W[k][n] + b[n], n padded to 16.
// fp32 WMMA 16x16x4: memory-bound (100 MB of hs streamed), fp32 keeps precision.
// ---------------------------------------------------------------------------
__global__ __launch_bounds__(256) void emis_kernel(const float* __restrict__ hs,
                                                   const float* __restrict__ W,
                                                   const float* __restrict__ bias,
                                                   float* __restrict__ epad) {
    // W transposed + zero-padded in LDS: WL[n*Hq + k] = (n<9) ? W[k][n] : 0
    __shared__ float WL[LP * Hq];   // 48 KB
    int tid = threadIdx.x;
    for (int idx = tid; idx < LP * Hq; idx += 256) {
        int n = idx / Hq, k = idx % Hq;
        WL[idx] = (n < Lq) ? W[k * Lq + n] : 0.0f;
    }
    __syncthreads();

    int wave = tid >> 5;            // 8 waves per block
    int lane = tid & 31;
    int n    = lane & 15;           // output column (and B-matrix lane)
    int hi   = lane >> 4;           // half-wave selects K pair / row offset
    int m0   = blockIdx.x * 128 + wave * 16;

    const float* arow = hs + (size_t)(m0 + n) * Hq + 2 * hi; // A: row m0+(lane&15)
    const float* bcol = WL + n * Hq + 2 * hi;                // B: col n

    v8f c = {};
    #pragma unroll 4
    for (int k0 = 0; k0 < Hq; k0 += 4) {
        v2f a  = *(const v2f*)(arow + k0);   // A[m][k0+2hi .. +1]
        v2f bb = *(const v2f*)(bcol + k0);   // B[k0+2hi .. +1][n]
        c = __builtin_amdgcn_wmma_f32_16x16x4_f32(
                /*neg_a=*/false, a, /*neg_b=*/false, bb,
                /*c_mod=*/(short)0, c, /*reuse_a=*/false, /*reuse_b=*/false);
    }

    float bn = (n < Lq) ? bias[n] : 0.0f;
    int r0 = m0 + hi * 8;           // C layout: vgpr i -> row r0+i, col n
    #pragma unroll
    for (int i = 0; i < 8; ++i) {
        epad[(size_t)(r0 + i) * LP + n] = c[i] + bn;
    }
}

// ---------------------------------------------------------------------------
// Kernel 2: per-batch CRF numerator + forward (logsumexp) scan. One wave/batch.
// Lanes 0..8 carry alpha[tag]; alpha exchanged via LDS (same-wave DS in-order;
// 1-wave workgroup barriers are S_NOP per ISA).
// ---------------------------------------------------------------------------
__global__ __launch_bounds__(32) void crf_kernel(const float* __restrict__ epad,
                                                 const int*   __restrict__ amask,
                                                 const int*   __restrict__ labels,
                                                 const float* __restrict__ start_trans,
                                                 const float* __restrict__ end_trans,
                                                 const float* __restrict__ trans,
                                                 float* __restrict__ out_b) {
    int b    = blockIdx.x;
    int lane = threadIdx.x;
    const int*   lab = labels + b * Sq;
    const int*   msk = amask  + b * Sq;
    const float* em  = epad + (size_t)b * Sq * LP;

    // ---- numerator: lanes stride over sequence positions ----
    float partial = 0.0f;
    int   cnt     = 0;
    for (int s = lane; s < Sq; s += 32) {
        int   l  = lab[s];
        float et = em[(size_t)s * LP + l];
        if (s == 0) {
            partial += start_trans[l] + et;
        } else {
            int   p  = lab[s - 1];
            float mf = (float)msk[s];
            partial += (trans[p * Lq + l] + et) * mf;
        }
        cnt += msk[s];
    }
    #pragma unroll
    for (int off = 16; off > 0; off >>= 1) {
        partial += __shfl_down(partial, off, 32);
        cnt     += __shfl_down(cnt,     off, 32);
    }
    cnt = __shfl(cnt, 0, 32);       // broadcast valid length
    float num = 0.0f;
    if (lane == 0) {
        int lt = lab[cnt - 1];
        num = partial + end_trans[lt];
    }

    // ---- forward scan ----
    __shared__ float aSh[16];
    __shared__ float red[16];
    float Tcol[Lq];
    float alpha = 0.0f;
    if (lane < Lq) {
        #pragma unroll
        for (int i = 0; i < Lq; ++i) Tcol[i] = trans[i * Lq + lane];
        alpha = start_trans[lane] + em[lane];
        aSh[lane] = alpha;
    }
    __syncthreads();
    for (int s = 1; s < Sq; ++s) {
        if (lane < Lq) {
            float t[Lq];
            float mx = -INFINITY;
            #pragma unroll
            for (int i = 0; i < Lq; ++i) {
                t[i] = aSh[i] + Tcol[i];
                mx = fmaxf(mx, t[i]);
            }
            float ss = 0.0f;
            #pragma unroll
            for (int i = 0; i < Lq; ++i) ss += __expf(t[i] - mx);
            float anew = em[(size_t)s * LP + lane] + mx + __logf(ss);
            alpha = msk[s] ? anew : alpha;
        }
        __syncthreads();            // S_NOP in 1-wave WG; compiler memory fence
        if (lane < Lq) aSh[lane] = alpha;
        __syncthreads();
    }

    if (lane < Lq) red[lane] = alpha + end_trans[lane];
    __syncthreads();
    if (lane == 0) {
        float mx = red[0];
        for (int i = 1; i < Lq; ++i) mx = fmaxf(mx, red[i]);
        float ss = 0.0f;
        for (int i = 0; i < Lq; ++i) ss += __expf(red[i] - mx);
        float denom = mx + __logf(ss);
        out_b[b] = denom - num;
    }
}

// ---------------------------------------------------------------------------
// Kernel 3: deterministic reduction of 64 per-batch values -> scalar.
// ---------------------------------------------------------------------------
__global__ __launch_bounds__(64) void sum_kernel(const float* __restrict__ out_b,
                                                 float* __restrict__ out) {
    __shared__ float sh[64];
    sh[threadIdx.x] = out_b[threadIdx.x];
    __syncthreads();
    if (threadIdx.x == 0) {
        float s = 0.0f;
        for (int i = 0; i < 64; ++i) s += sh[i];
        out[0] = s;
    }
}

extern "C" void kernel_launch(void* const* d_in, const int* in_sizes, int n_in,
                              void* d_out, int out_size, void* d_ws, size_t ws_size,
                              hipStream_t stream) {
    const float* hs     = (const float*)d_in[0];  // (64,512,768) f32
    const int*   amask  = (const int*)  d_in[1];  // (64,512) i32
    const int*   labels = (const int*)  d_in[2];  // (64,512) i32
    const float* W      = (const float*)d_in[3];  // (768,9)  f32
    const float* bias   = (const float*)d_in[4];  // (9,)     f32
    const float* st     = (const float*)d_in[5];  // (9,)
    const float* et     = (const float*)d_in[6];  // (9,)
    const float* tr     = (const float*)d_in[7];  // (9,9)

    float* epad  = (float*)d_ws;                  // 32768*16 f32 = 2 MB
    float* out_b = epad + (size_t)Mq * LP;        // 64 f32

    emis_kernel<<<Mq / 128, 256, 0, stream>>>(hs, W, bias, epad);
    crf_kernel<<<Bq, 32, 0, stream>>>(epad, amask, labels, st, et, tr, out_b);
    sum_kernel<<<1, 64, 0, stream>>>(out_b, (float*)d_out);
}